// TritonLinearET_34626026340513
// MI455X (gfx1250) — compile-verified
//
#include <hip/hip_runtime.h>
#include <math.h>

typedef float v2f __attribute__((ext_vector_type(2)));
typedef float v8f __attribute__((ext_vector_type(8)));
typedef int   v4i __attribute__((vector_size(16)));

#define TAU_F 2.053748910631823f

constexpr int BM  = 128;   // workgroup tile rows (of x)
constexpr int BN  = 64;    // workgroup tile cols (rows of W)
constexpr int KB  = 32;    // k-block (first 16 of block 0 = screening window K0)
constexpr int LST = 36;    // padded LDS row stride in floats (144B: 16B-aligned, conflict-free)

// ---- async global -> LDS copy (CDNA5 GLOBAL_LOAD_ASYNC_TO_LDS_B128) ----
__device__ __forceinline__ void async_copy_b128(const float* g, float* l) {
#if __has_builtin(__builtin_amdgcn_global_load_async_to_lds_b128)
    typedef __attribute__((address_space(1))) v4i GA;   // global v4i
    typedef __attribute__((address_space(3))) v4i LA;   // LDS v4i
    __builtin_amdgcn_global_load_async_to_lds_b128(
        (GA*)(v4i*)(float*)g,   // drop const, reinterpret, addrspacecast -> as(1)
        (LA*)(v4i*)l,           // reinterpret, addrspacecast -> as(3)
        0, 0);
#else
    const unsigned lds_off = (unsigned)(unsigned long long)
        (__attribute__((address_space(3))) void*)l;
    asm volatile("global_load_async_to_lds_b128 %0, %1, off"
                 :: "v"(lds_off), "v"(g) : "memory");
#endif
}

__device__ __forceinline__ void wait_async0() {
#if __has_builtin(__builtin_amdgcn_s_wait_asynccnt)
    __builtin_amdgcn_s_wait_asynccnt(0);
#else
    asm volatile("s_wait_asynccnt 0x0" ::: "memory");
#endif
}

__global__ __launch_bounds__(256)
void fused_screen_gemm(const float* __restrict__ x,
                       const float* __restrict__ W,
                       const float* __restrict__ bias,
                       float* __restrict__ out,
                       int B, int K, int N)
{
    __shared__ float xs[2][BM * LST];
    __shared__ float ws[2][BN * LST];

    const int tid  = threadIdx.x;
    const int lane = tid & 31;
    const int wave = tid >> 5;
    const int lrow = lane & 15;      // row of A / col of B,D within fragment
    const int half = lane >> 4;      // K-pair select (A/B) / +8 rows (D)

    const int waveM = wave & 3;      // 4 waves along M
    const int waveN = wave >> 2;     // 2 waves along N
    const int bm = blockIdx.y * BM;
    const int bn = blockIdx.x * BN;

    // ---- async staging: 8 threads x float4 = 128B per row, no VGPR staging ----
    const int ldr  = tid >> 3;           // 0..31
    const int ldc4 = (tid & 7) * 4;      // 0,4,...,28
    auto astage = [&](int k0, int buf) {
        #pragma unroll
        for (int p = 0; p < 4; ++p)
            async_copy_b128(&x[(size_t)(bm + ldr + p * 32) * K + k0 + ldc4],
                            &xs[buf][(ldr + p * 32) * LST + ldc4]);
        #pragma unroll
        for (int p = 0; p < 2; ++p)
            async_copy_b128(&W[(size_t)(bn + ldr + p * 32) * K + k0 + ldc4],
                            &ws[buf][(ldr + p * 32) * LST + ldc4]);
    };

    // LDS fragment loads (A from xs, B from ws)
    auto lda = [&](int buf, int mf, int kk) -> v2f {
        return *(const v2f*)&xs[buf][(waveM * 32 + mf * 16 + lrow) * LST + kk + 2 * half];
    };
    auto ldb = [&](int buf, int nf, int kk) -> v2f {
        return *(const v2f*)&ws[buf][(waveN * 32 + nf * 16 + lrow) * LST + kk + 2 * half];
    };

    v8f acc[2][2];
    #pragma unroll
    for (int mf = 0; mf < 2; ++mf)
        #pragma unroll
        for (int nf = 0; nf < 2; ++nf)
            acc[mf][nf] = (v8f)(0.0f);

    auto mma4 = [&](int buf, int kk) {
        v2f a[2], b[2];
        #pragma unroll
        for (int mf = 0; mf < 2; ++mf) a[mf] = lda(buf, mf, kk);
        #pragma unroll
        for (int nf = 0; nf < 2; ++nf) b[nf] = ldb(buf, nf, kk);
        #pragma unroll
        for (int mf = 0; mf < 2; ++mf)
            #pragma unroll
            for (int nf = 0; nf < 2; ++nf)
                acc[mf][nf] = __builtin_amdgcn_wmma_f32_16x16x4_f32(
                    false, a[mf], false, b[nf], (short)0, acc[mf][nf], false, false);
    };

    // ================= prologue: block 0 with screening test =================
    astage(0, 0);
    wait_async0();
    __syncthreads();

    astage(KB, 1);                  // block 1 in flight during block-0 compute

    // First 16 k's: acc accumulates y1; s2 accumulates (x^2)(w^2) via squared WMMAs.
    unsigned masks[2][2];
    {
        v8f s2[2][2];
        #pragma unroll
        for (int mf = 0; mf < 2; ++mf)
            #pragma unroll
            for (int nf = 0; nf < 2; ++nf)
                s2[mf][nf] = (v8f)(0.0f);

        #pragma unroll
        for (int kk = 0; kk < 16; kk += 4) {
            v2f a[2], b[2];
            #pragma unroll
            for (int mf = 0; mf < 2; ++mf) a[mf] = lda(0, mf, kk);
            #pragma unroll
            for (int nf = 0; nf < 2; ++nf) b[nf] = ldb(0, nf, kk);
            #pragma unroll
            for (int mf = 0; mf < 2; ++mf)
                #pragma unroll
                for (int nf = 0; nf < 2; ++nf) {
                    acc[mf][nf] = __builtin_amdgcn_wmma_f32_16x16x4_f32(
                        false, a[mf], false, b[nf], (short)0, acc[mf][nf], false, false);
                    s2[mf][nf] = __builtin_amdgcn_wmma_f32_16x16x4_f32(
                        false, a[mf] * a[mf], false, b[nf] * b[nf], (short)0, s2[mf][nf], false, false);
                }
        }

        #pragma unroll
        for (int mf = 0; mf < 2; ++mf)
            #pragma unroll
            for (int nf = 0; nf < 2; ++nf) {
                unsigned m = 0;
                #pragma unroll
                for (int r = 0; r < 8; ++r) {
                    const float y1v = acc[mf][nf][r];
                    const float s2v = s2[mf][nf][r];
                    // tstat < TAU  <=>  |y1| < TAU*sqrt(s2/16); s2==0 -> false (matches NaN<TAU)
                    if (fabsf(y1v) < TAU_F * sqrtf(s2v * 0.0625f)) m |= (1u << r);
                }
                masks[mf][nf] = m;
            }
    }

    // rest of block 0
    #pragma unroll
    for (int kk = 16; kk < KB; kk += 4) mma4(0, kk);

    wait_async0();
    __syncthreads();

    // ================= steady-state pipeline =================
    int cur = 1;
    for (int k0 = 2 * KB; k0 < K; k0 += KB) {
        astage(k0, cur ^ 1);                                 // async engine fills next buf...
        #pragma unroll
        for (int kk = 0; kk < KB; kk += 4) mma4(cur, kk);    // ...behind 32 WMMAs
        wait_async0();
        __syncthreads();
        cur ^= 1;
    }
    // last block
    #pragma unroll
    for (int kk = 0; kk < KB; kk += 4) mma4(cur, kk);

    // ================= epilogue: mask, bias, store =================
    #pragma unroll
    for (int nf = 0; nf < 2; ++nf) {
        const int ncolg = bn + waveN * 32 + nf * 16 + lrow;
        const float bv = bias[ncolg];
        #pragma unroll
        for (int mf = 0; mf < 2; ++mf) {
            const unsigned m = masks[mf][nf];
            #pragma unroll
            for (int r = 0; r < 8; ++r) {
                const int grow = bm + waveM * 32 + mf * 16 + r + 8 * half;
                const float v = ((m >> r) & 1u) ? 0.0f : (acc[mf][nf][r] + bv);
                out[(size_t)grow * N + ncolg] = v;
            }
        }
    }
}

extern "C" void kernel_launch(void* const* d_in, const int* in_sizes, int n_in,
                              void* d_out, int out_size, void* d_ws, size_t ws_size,
                              hipStream_t stream) {
    const float* x    = (const float*)d_in[0];
    const float* W    = (const float*)d_in[1];
    const float* bias = (const float*)d_in[2];
    float* out        = (float*)d_out;

    const int DOUT = in_sizes[2];                 // 4096
    const int DIN  = in_sizes[1] / DOUT;          // 4096
    const int B    = in_sizes[0] / DIN;           // 8192

    // blockIdx.x sweeps N-tiles for a fixed 128-row x-stripe -> the stripe (2MB)
    // and all of W (64MB) stay resident in the 192MB L2.
    dim3 grid(DOUT / BN, B / BM);
    fused_screen_gemm<<<grid, 256, 0, stream>>>(x, W, bias, out, B, DIN, DOUT);
}